// Mask_28226525070050
// MI455X (gfx1250) — compile-verified
//
#include <hip/hip_runtime.h>
#include <hip/hip_bf16.h>
#include <stdint.h>

#define DIMN 192
#define TW 32
#define THH 8
#define HALO_W 34           // TW+2
#define HALO_H 10           // THH+2
#define PLANE (HALO_W*HALO_H)   // 340 floats per halo plane
#define NSLOT 4

// ---- CDNA5 async global->LDS copy (ASYNCcnt path), inline asm (portable) ----
__device__ __forceinline__ void async_load_to_lds_b32(uint32_t lds_byte_addr,
                                                      uint64_t global_addr) {
    // VDST = VGPR holding LDS byte address, VADDR = 64-bit global address
    asm volatile("global_load_async_to_lds_b32 %0, %1, off"
                 :: "v"(lds_byte_addr), "v"(global_addr)
                 : "memory");
}
__device__ __forceinline__ void wait_async_le2() {
    asm volatile("s_wait_asynccnt 0x2" ::: "memory");
}
__device__ __forceinline__ void wait_async_0() {
    asm volatile("s_wait_asynccnt 0x0" ::: "memory");
}

__global__ __launch_bounds__(256)
void Mask_28226525070050_kernel(const float* __restrict__ m,
                                float* __restrict__ out) {
    __shared__ float tile[NSLOT * PLANE];

    const int tx  = threadIdx.x;            // 0..31 (w)
    const int ty  = threadIdx.y;            // 0..7  (h)
    const int tid = ty * TW + tx;           // 0..255
    const int w0  = blockIdx.x * TW;
    const int h0  = blockIdx.y * THH;
    const int b   = blockIdx.z;
    const int w   = w0 + tx;
    const int h   = h0 + ty;

    const uint64_t base =
        (uint64_t)(uintptr_t)m + (uint64_t)b * (uint64_t)(DIMN*DIMN*DIMN) * 4ull;

    // Two halo cells per thread per plane: e1 in [0,255], e2 in [84,339].
    // Cells 84..255 are loaded twice from identical addresses (benign, keeps
    // EXEC full and ASYNCcnt deterministic: exactly 2 issues/wave/plane).
    const int e1 = tid;
    const int e2 = tid + (PLANE - 256);     // +84
    const int hy1 = e1 / HALO_W, wx1 = e1 - hy1 * HALO_W;
    const int hy2 = e2 / HALO_W, wx2 = e2 - hy2 * HALO_W;
    const int gh1 = min(max(h0 - 1 + hy1, 0), DIMN - 1);
    const int gw1 = min(max(w0 - 1 + wx1, 0), DIMN - 1);
    const int gh2 = min(max(h0 - 1 + hy2, 0), DIMN - 1);
    const int gw2 = min(max(w0 - 1 + wx2, 0), DIMN - 1);
    const uint64_t off1 = (uint64_t)(gh1 * DIMN + gw1) * 4ull;
    const uint64_t off2 = (uint64_t)(gh2 * DIMN + gw2) * 4ull;
    // LDS byte addresses (low 32 bits of generic shared pointer = LDS offset)
    const uint32_t lbase = (uint32_t)(uintptr_t)&tile[0];
    const uint32_t l1 = lbase + (uint32_t)(e1 * 4);
    const uint32_t l2 = lbase + (uint32_t)(e2 * 4);

    // Boundary masks (float); C = #out-of-bounds face neighbors = 6 - sum(masks)
    const float mh0 = (h > 0)        ? 1.f : 0.f;
    const float mh1 = (h < DIMN - 1) ? 1.f : 0.f;
    const float mw0 = (w > 0)        ? 1.f : 0.f;
    const float mw1 = (w < DIMN - 1) ? 1.f : 0.f;
    const float msum_hw = mh0 + mh1 + mw0 + mw1;

    const int cc = (ty + 1) * HALO_W + (tx + 1);

    // Prologue: kick off planes 0 and 1 (slots 0,1). Slot for plane p = p & 3.
    {
        const uint64_t pb0 = base;
        const uint64_t pb1 = base + (uint64_t)(DIMN*DIMN) * 4ull;
        async_load_to_lds_b32(l1 + 0 * PLANE * 4, pb0 + off1);
        async_load_to_lds_b32(l2 + 0 * PLANE * 4, pb0 + off2);
        async_load_to_lds_b32(l1 + 1 * PLANE * 4, pb1 + off1);
        async_load_to_lds_b32(l2 + 1 * PLANE * 4, pb1 + off2);
    }

    float* outb = out + (size_t)b * DIMN * DIMN * DIMN + (size_t)h * DIMN + w;

    for (int d = 0; d < DIMN; ++d) {
        // Issue plane d+2 into slot (d+2)&3 (its previous tenant, plane d-2,
        // was last read by iteration d-1, protected by trailing barrier).
        if (d + 2 < DIMN) {
            const int p = d + 2;
            const uint64_t pb = base + (uint64_t)p * (uint64_t)(DIMN*DIMN) * 4ull;
            const uint32_t ls = (uint32_t)((p & 3) * PLANE * 4);
            async_load_to_lds_b32(l1 + ls, pb + off1);
            async_load_to_lds_b32(l2 + ls, pb + off2);
            wait_async_le2();   // outstanding: plane d+2 only -> d+1 resident
        } else {
            wait_async_0();     // drain: plane d+1 resident
        }
        __syncthreads();        // cross-wave visibility of planes <= d+1

        const int s0 = (d + 3) & 3;   // plane d-1
        const int s1 = d & 3;         // plane d
        const int s2 = (d + 1) & 3;   // plane d+1

        const float c  = tile[s1 * PLANE + cc];
        const float up = tile[s1 * PLANE + cc - HALO_W];
        const float dn = tile[s1 * PLANE + cc + HALO_W];
        const float lf = tile[s1 * PLANE + cc - 1];
        const float rt = tile[s1 * PLANE + cc + 1];
        // d-halo slots may hold uninitialized LDS at d==0 / d==191: use selects
        // (mask*garbage could be NaN-poisoned), condition is wave-uniform.
        const float fr = (d > 0)        ? tile[s0 * PLANE + cc] : 0.f;
        const float bk = (d < DIMN - 1) ? tile[s2 * PLANE + cc] : 0.f;

        const float md0 = (d > 0)        ? 1.f : 0.f;
        const float md1 = (d < DIMN - 1) ? 1.f : 0.f;

        // L = conv(m) at this point; conv(1-m) = C - L with C = #OOB neighbors
        const float L = 6.f * c - (mh0 * up + mh1 * dn + mw0 * lf + mw1 * rt + fr + bk);
        const float C = 6.f - (msum_hw + md0 + md1);

        const float val = ((L >= 0.f) || ((C - L) >= 0.5f)) ? 1.f : 0.f;
        __builtin_nontemporal_store(val, outb + (size_t)d * (DIMN * DIMN));

        __syncthreads();        // everyone done reading before slot reuse
    }
}

extern "C" void kernel_launch(void* const* d_in, const int* in_sizes, int n_in,
                              void* d_out, int out_size, void* d_ws, size_t ws_size,
                              hipStream_t stream) {
    const float* m = (const float*)d_in[0];
    // d_in[1] is the 3x3x3 Laplacian kernel; it is a fixed constant in the
    // reference, folded analytically into the stencil above.
    float* out = (float*)d_out;
    const int B = in_sizes[0] / (DIMN * DIMN * DIMN);   // = 2
    dim3 grid(DIMN / TW, DIMN / THH, B);
    dim3 block(TW, THH, 1);
    hipLaunchKernelGGL(Mask_28226525070050_kernel, grid, block, 0, stream, m, out);
}